// GDeltaAttention_6614249636015
// MI455X (gfx1250) — compile-verified
//
#include <hip/hip_runtime.h>

// Problem constants (from reference): B=8, N=128, D=128, IN=1024, K=32
#define Bv   8
#define Nv   128
#define Dv   128
#define INv  1024
#define Kv   32
#define COLS (Nv * 2 * Dv)   // 32768 gate columns

typedef __attribute__((ext_vector_type(2))) float v2f;
typedef __attribute__((ext_vector_type(4))) float v4f;
typedef __attribute__((ext_vector_type(8))) float v8f;

static __device__ __forceinline__ float sigmoidf(float x) {
    return __fdividef(1.0f, 1.0f + __expf(-x));
}

// ---------------------------------------------------------------------------
// Kernel 1: gates = x_t @ W_gate.T + b_gate ; ab = sigmoid(sigmoid(gates))
//
// Pure-fp32 WMMA (V_WMMA_F32_16X16X4_F32): no precision loss, no convert VALU.
// One 16x16 tile per wave; C rows 0..7 are the 8 batches.
//
// A rows 8..15 are NOT zero-padded: C[m] depends only on A[m], so garbage in
// A rows 8..15 only pollutes C rows 8..15, which are discarded (WMMA raises
// no exceptions). We clamp the row index (m & 7) so loads stay in-bounds --
// this keeps the inner loop branch-free (no exec-mask save/restore, no
// zero-fills), letting the two b128 loads clause together.
//
// f32 16x16x4 wave32 fragment layouts (CDNA5 ISA 7.12.2):
//   A (16x4):  lane l: M=l&15, hi=l>>4; VGPR0 -> slice pos 2*hi, VGPR1 -> pos 2*hi+1
//   B (4x16):  lane l: N=l&15, hi=l>>4; VGPR0 -> slice pos 2*hi, VGPR1 -> pos 2*hi+1
//   C/D:       VGPR i, lanes 0-15: M=i, N=lane   (lanes 16-31: M=i+8, discarded)
//
// K-slice remap: accumulation order over K is free, so per 8 K-values hi=0
// lanes own {k..k+3} and hi=1 lanes own {k+4..k+7}; one contiguous float4
// per lane then feeds TWO wmmas:
//   wmma #1 slice = {k,k+1 | k+4,k+5},  wmma #2 slice = {k+2,k+3 | k+6,k+7}
// (identical mapping for A and B, so products pair up correctly).
// ---------------------------------------------------------------------------
__global__ __launch_bounds__(256) void gate_gemm_wmma(
    const float* __restrict__ x_t,     // [8, 1024]
    const float* __restrict__ W_gate,  // [32768, 1024] row-major
    const float* __restrict__ b_gate,  // [32768]
    float* __restrict__ ab)            // [8, 32768] sigmoid(sigmoid(gates))
{
    const int lane = threadIdx.x & 31;
    const int wid  = threadIdx.x >> 5;
    const int tile = blockIdx.x * 8 + wid;   // 0..2047
    const int col0 = tile * 16;
    const int m    = lane & 15;              // A row (batch) / B column-in-tile
    const int hi   = lane >> 4;

    // Clamped row: lanes m>=8 re-read a valid row; their results are discarded.
    const float* arow = x_t + (size_t)(m & 7) * INv;
    const float* brow = W_gate + ((size_t)(col0 + m)) * INv; // B[k,n] = W_gate[col0+n, k]

    v8f c = {};

    #pragma unroll 4
    for (int k8 = 0; k8 < INv; k8 += 8) {
        const int base = k8 + 4 * hi;

        v4f fa = *(const v4f*)(arow + base);                       // L2/L0 hot
        v4f fb = __builtin_nontemporal_load((const v4f*)(brow + base)); // streamed once

        v2f a0 = { fa.x, fa.y }, a1 = { fa.z, fa.w };
        v2f b0 = { fb.x, fb.y }, b1 = { fb.z, fb.w };

        // 8 args: (neg_a, A, neg_b, B, c_mod, C, reuse_a, reuse_b)
        c = __builtin_amdgcn_wmma_f32_16x16x4_f32(
                false, a0, false, b0, (short)0, c, false, false);
        c = __builtin_amdgcn_wmma_f32_16x16x4_f32(
                false, a1, false, b1, (short)0, c, false, false);
    }

    // Lanes 0-15 hold M=0..7 (the real batches) across c[0..7].
    if (hi == 0) {
        const int   colg = col0 + lane;
        const float bg   = b_gate[colg];
        #pragma unroll
        for (int i = 0; i < Bv; ++i) {
            float g = c[i] + bg;
            ab[(size_t)i * COLS + colg] = sigmoidf(sigmoidf(g));
        }
    }
}

// ---------------------------------------------------------------------------
// Kernel 2: per (b,n) slice of M.
//   inactive: out = M                    (NT float4 streaming copy + prefetch)
//   active:   recall[e] = sum_d M_k[d]*M[d,e]
//             bracket[e] = beta[e]*M_v[e] - recall[e]*alpha[e]
//             out[d,e]  = alpha[e]*M[d,e] + M_k[d]*bracket[e]
// Grid: B*N = 1024 blocks, 256 threads (2 d-rows x 128 e-columns).
// ---------------------------------------------------------------------------
__global__ __launch_bounds__(256) void delta_update(
    const float* __restrict__ M,    // [B,N,D,D]
    const float* __restrict__ M_k,  // [B,N,D]
    const float* __restrict__ M_v,  // [B,N,D]
    const int*   __restrict__ idx,  // [B,K]
    const float* __restrict__ ab,   // [B, N*2D]
    float* __restrict__ out)        // [B,N,D,D]
{
    const int bn  = blockIdx.x;
    const int b   = bn >> 7;
    const int n   = bn & 127;
    const int tid = threadIdx.x;

    bool active = false;
    #pragma unroll
    for (int j = 0; j < Kv; ++j) active |= (idx[b * Kv + j] == n);

    const size_t mbase = (size_t)bn * (Dv * Dv);

    if (!active) {
        const v4f* src = (const v4f*)(M + mbase);
        v4f*       dst = (v4f*)(out + mbase);
        #pragma unroll
        for (int i = 0; i < 16; ++i) {           // 4096 float4 / 256 threads
            if (i < 14)
                __builtin_prefetch((const void*)(src + tid + (i + 2) * 256), 0, 0);
            v4f v = __builtin_nontemporal_load(src + tid + i * 256);
            __builtin_nontemporal_store(v, dst + tid + i * 256);
        }
        return;
    }

    __shared__ float sMk[Dv];
    __shared__ float sAl[Dv];
    __shared__ float sBr[Dv];
    __shared__ float sPart[2][Dv];

    if (tid < Dv) {
        sMk[tid] = M_k[(size_t)bn * Dv + tid];
        sAl[tid] = ab[(size_t)b * COLS + (size_t)n * (2 * Dv) + tid];       // alpha
    }
    __syncthreads();

    const int r = tid >> 7;    // 0 or 1 : d-parity
    const int e = tid & 127;   // column

    float acc = 0.0f;
    #pragma unroll 4
    for (int d = r; d < Dv; d += 2)
        acc += sMk[d] * M[mbase + (size_t)d * Dv + e];
    sPart[r][e] = acc;
    __syncthreads();

    if (tid < Dv) {
        float recall = sPart[0][tid] + sPart[1][tid];
        float beta   = ab[(size_t)b * COLS + (size_t)n * (2 * Dv) + Dv + tid];
        float mv     = M_v[(size_t)bn * Dv + tid];
        sBr[tid]     = beta * mv - recall * sAl[tid];
    }
    __syncthreads();

    #pragma unroll 4
    for (int d = r; d < Dv; d += 2) {
        float mk = sMk[d];
        float mval = M[mbase + (size_t)d * Dv + e];   // L2-hot (same 64KB tile)
        out[mbase + (size_t)d * Dv + e] = sAl[e] * mval + mk * sBr[e];
    }
}

// ---------------------------------------------------------------------------
extern "C" void kernel_launch(void* const* d_in, const int* in_sizes, int n_in,
                              void* d_out, int out_size, void* d_ws, size_t ws_size,
                              hipStream_t stream) {
    const float* M      = (const float*)d_in[0];
    const float* M_k    = (const float*)d_in[1];
    const float* M_v    = (const float*)d_in[2];
    const int*   idx    = (const int*)d_in[3];
    const float* x_t    = (const float*)d_in[4];
    const float* W_gate = (const float*)d_in[5];
    const float* b_gate = (const float*)d_in[6];
    float*       out    = (float*)d_out;
    float*       ab     = (float*)d_ws;   // 8 * 32768 floats = 1 MB scratch

    // 2048 column tiles, 8 waves (one tile each) per 256-thread block.
    gate_gemm_wmma<<<256, 256, 0, stream>>>(x_t, W_gate, b_gate, ab);
    // One block per (b, n) slice.
    delta_update<<<Bv * Nv, 256, 0, stream>>>(M, M_k, M_v, idx, ab, out);
}